// RelativePositionEncoding_88175678587717
// MI455X (gfx1250) — compile-verified
//
#include <hip/hip_runtime.h>
#include <cstdint>

// ---- problem constants (from reference) ----
#define R_TOKEN_MAX 32
#define R_CHAIN_MAX 2
#define C_PAIR      128
#define D_TOK       (2 * R_TOKEN_MAX + 2)     // 66
#define D_CHAIN     (2 * R_CHAIN_MAX + 2)     // 6
#define D_IN        (2 * D_TOK + 1 + D_CHAIN) // 139
#define N_MAX       1024
#define C4          (C_PAIR / 4)              // 32 float4 per row
#define NW_FLOATS   (D_IN * C_PAIR)           // 17792 floats = 71168 B
#define NW_F4       (NW_FLOATS / 4)           // 4448

typedef float    f32x4 __attribute__((ext_vector_type(4)));
typedef uint32_t u32x4 __attribute__((ext_vector_type(4)));
typedef int      i32x4 __attribute__((ext_vector_type(4)));
typedef int      i32x8 __attribute__((ext_vector_type(8)));

__device__ __forceinline__ int iclamp(int v, int lo, int hi) {
  return v < lo ? lo : (v > hi ? hi : v);
}

// One block per output row i. 256 threads = 8 waves; each wave processes one
// pair (i, j) per iteration, lane l covering channels [4l, 4l+4).
__global__ __launch_bounds__(256) void relpos_kernel(
    const int* __restrict__ res, const int* __restrict__ tok,
    const int* __restrict__ asym, const int* __restrict__ ent,
    const int* __restrict__ sym, const float* __restrict__ W,
    float* __restrict__ out, int N) {
  __shared__ float sW[NW_FLOATS];       // full W table: 139 x 128 f32
  __shared__ int   sIdx[5 * N_MAX];     // res | tok | asym | ent | sym

  const int tid  = threadIdx.x;
  const int i    = blockIdx.x;
  const int lane = tid & 31;
  const int wave = tid >> 5;

  // Stage the 5 per-token index arrays into LDS (N <= 1024).
  for (int k = tid; k < N; k += 256) {
    sIdx[k]             = res[k];
    sIdx[N_MAX + k]     = tok[k];
    sIdx[2 * N_MAX + k] = asym[k];
    sIdx[3 * N_MAX + k] = ent[k];
    sIdx[4 * N_MAX + k] = sym[k];
  }

#if defined(__HIP_DEVICE_COMPILE__) && __has_builtin(__builtin_amdgcn_tensor_load_to_lds)
  // Tensor Data Mover: DMA the whole W table (2-D tile, 139 rows x 128 f32,
  // dim0 stride 128 -> fully contiguous) from global into LDS. One wave issues
  // the descriptor; it waits TENSORcnt==0, then everyone syncs at the barrier.
  if (tid < 32) {
    const uint64_t g = (uint64_t)(uintptr_t)W;
    const uint32_t lds_base = (uint32_t)(uintptr_t)(&sW[0]); // low 32b of flat LDS addr = LDS offset

    u32x4 g0;
    g0.x = 1u;                                   // count=1, is_restore=0, gather off
    g0.y = lds_base;                             // lds_addr (bytes)
    g0.z = (uint32_t)(g & 0xFFFFFFFFu);          // global_addr[31:0]
    g0.w = (uint32_t)((g >> 32) & 0x1FFFFFFu)    // global_addr[56:32]
         | (2u << 30);                           // type=2 ("image")

    i32x8 g1;
    g1.s0 = (int)(2u << 16);                     // workgroup_mask=0, data_size=2 (4 B)
    g1.s1 = (int)((uint32_t)C_PAIR << 16);       // tensor_dim0 = 128 (atomic barrier addr = 0)
    g1.s2 = (int)((uint32_t)D_IN << 16);         // tensor_dim0 hi=0 | tensor_dim1 = 139
    g1.s3 = (int)((uint32_t)C_PAIR << 16);       // tensor_dim1 hi=0 | tile_dim0 = 128
    g1.s4 = D_IN;                                // tile_dim1 = 139, tile_dim2 = 0 (2-D)
    g1.s5 = C_PAIR;                              // tensor_dim0_stride = 128 (lo 32)
    g1.s6 = 0;                                   // stride hi / dim1_stride (unused, 2-D)
    g1.s7 = 0;

    i32x4 g2 = {0, 0, 0, 0};                     // dims 2/3 unused (tile_dim2/3 = 0)
    i32x4 g3 = {0, 0, 0, 0};
    i32x8 g4 = {0, 0, 0, 0, 0, 0, 0, 0};         // extra descriptor group (unused)

    __builtin_amdgcn_tensor_load_to_lds(g0, g1, g2, g3, g4, 0);
    __builtin_amdgcn_s_wait_tensorcnt(0);
  }
#else
  // Fallback: cooperative b128 copy of W into LDS.
  for (int k = tid; k < NW_F4; k += 256) {
    ((f32x4*)sW)[k] = ((const f32x4*)W)[k];
  }
#endif
  __syncthreads();

  // Row-i scalars (wave-uniform for whole block).
  const int ri = sIdx[i];
  const int ti = sIdx[N_MAX + i];
  const int ai = sIdx[2 * N_MAX + i];
  const int ei = sIdx[3 * N_MAX + i];
  const int si = sIdx[4 * N_MAX + i];

  const f32x4* sW4 = (const f32x4*)sW;
  const f32x4  we  = sW4[(2 * D_TOK) * C4 + lane];  // w_ent chunk (row 132)

  for (int j = wave; j < N; j += 8) {
    const int rj = sIdx[j];
    const int tj = sIdx[N_MAX + j];
    const int aj = sIdx[2 * N_MAX + j];
    const int ej = sIdx[3 * N_MAX + j];
    const int sj = sIdx[4 * N_MAX + j];

    const bool same_chain   = (ai == aj);
    const bool same_residue = (ri == rj);

    const int ip = same_residue ? iclamp(ri - rj + R_TOKEN_MAX, 0, 2 * R_TOKEN_MAX)
                                : (2 * R_TOKEN_MAX + 1);                       // 0..65
    const int it = (same_residue && same_chain)
                     ? iclamp(ti - tj + R_TOKEN_MAX, 0, 2 * R_TOKEN_MAX)
                     : (2 * R_TOKEN_MAX + 1);                                  // 0..65
    const int ic = (!same_chain) ? iclamp(si - sj + R_CHAIN_MAX, 0, 2 * R_CHAIN_MAX)
                                 : (2 * R_CHAIN_MAX + 1);                      // 0..5

    // Three ds_load_b128 gathers from the LDS-resident tables.
    const f32x4 a = sW4[ip * C4 + lane];                       // Wp[idx_pos]
    const f32x4 b = sW4[(D_TOK + it) * C4 + lane];             // Wt[idx_tok]
    const f32x4 c = sW4[(2 * D_TOK + 1 + ic) * C4 + lane];     // Wc[idx_chain]

    f32x4 acc = a + b + c;
    if (ei == ej) acc += we;   // wave-uniform branch

    // Contiguous 512 B per wave, non-temporal (output stream >> L2).
    f32x4* dst = (f32x4*)(out + ((size_t)i * (size_t)N + (size_t)j) * C_PAIR) + lane;
    __builtin_nontemporal_store(acc, dst);
  }
}

extern "C" void kernel_launch(void* const* d_in, const int* in_sizes, int n_in,
                              void* d_out, int out_size, void* d_ws, size_t ws_size,
                              hipStream_t stream) {
  const int*   res  = (const int*)d_in[0];   // residue_index (B=1, N)
  const int*   tok  = (const int*)d_in[1];   // token_index
  const int*   asym = (const int*)d_in[2];   // asym_id
  const int*   ent  = (const int*)d_in[3];   // entity_id
  const int*   sym  = (const int*)d_in[4];   // sym_id
  const float* W    = (const float*)d_in[5]; // (139, 128) f32
  float*       out  = (float*)d_out;         // (N, N, 128) f32

  const int N = in_sizes[0];                 // B=1 -> flat count == N (1024)

  dim3 grid(N), block(256);
  relpos_kernel<<<grid, block, 0, stream>>>(res, tok, asym, ent, sym, W, out, N);
}